// AttentionPool_76235669504352
// MI455X (gfx1250) — compile-verified
//
#include <hip/hip_runtime.h>
#include <math.h>

typedef __attribute__((ext_vector_type(2))) float v2f;
typedef __attribute__((ext_vector_type(8))) float v8f;

constexpr int kD      = 512;            // feature dim
constexpr int kN      = 8192;           // rows per batch
constexpr int kB      = 16;             // batches
constexpr int kChunk  = 128;            // rows per block (8 waves * 16 rows)
constexpr int kNChunk = kN / kChunk;    // 64 chunks per batch
constexpr float kEps  = 1e-5f;
constexpr float kScale = 0.04419417382415922f;  // 1/sqrt(512)

// ws layout (floats):
//   [0, 512)                      u = qn * ln_w
//   [512]  sumU      [513]  qb = dot(qn, ln_b)
//   [1024, 1024 + 2*kB*kNChunk)   per-chunk (m, s)
//   [4096, 4096 + kB*kNChunk*kD)  per-chunk pooled partials   (~2.1 MB total)
constexpr int WS_SCAL = 512;
constexpr int WS_MS   = 1024;
constexpr int WS_P    = 4096;

// ---------------------------------------------------------------------------
// Kernel 1: layernorm the query, build u = qn*w, sumU, qb.  One tiny block.
// ---------------------------------------------------------------------------
__global__ __launch_bounds__(kD) void ap_prep(const float* __restrict__ q,
                                              const float* __restrict__ lw,
                                              const float* __restrict__ lb,
                                              float* __restrict__ ws) {
  __shared__ float ra[kD], rb[kD];
  const int t = threadIdx.x;
  const float qv = q[t];
  ra[t] = qv;
  rb[t] = qv * qv;
  __syncthreads();
  for (int s = kD / 2; s >= 1; s >>= 1) {
    if (t < s) { ra[t] += ra[t + s]; rb[t] += rb[t + s]; }
    __syncthreads();
  }
  const float mu  = ra[0] * (1.0f / kD);
  const float var = rb[0] * (1.0f / kD) - mu * mu;
  const float rs  = rsqrtf(var + kEps);
  __syncthreads();
  const float qn = (qv - mu) * rs * lw[t] + lb[t];
  const float u  = qn * lw[t];
  ws[t] = u;
  ra[t] = u;
  rb[t] = qn * lb[t];
  __syncthreads();
  for (int s = kD / 2; s >= 1; s >>= 1) {
    if (t < s) { ra[t] += ra[t + s]; rb[t] += rb[t + s]; }
    __syncthreads();
  }
  if (t == 0) { ws[WS_SCAL] = ra[0]; ws[WS_SCAL + 1] = rb[0]; }
}

// ---------------------------------------------------------------------------
// Kernel 2: one block per (batch, 128-row chunk).  Single HBM pass over x.
//   Phase 1 (per wave, 16 rows): three WMMA accumulators over the K loop:
//       cd = u^T   X   -> dot(u, x_row)      (A = broadcast u chunk)
//       cs = 1^T   X   -> sum(x_row)         (A = ones, loop-invariant)
//       cq = 1^T (X*X) -> sum(x_row^2)
//     B carries the x tile: B[k][n] = x[row n][k0+k]; every C row is
//     identical, so row 0 (VGPR0, lanes 0-15) gives all 16 per-row results.
//     Loop body is branch-free: global_load_b64 + ds_load_b64 (broadcast)
//     + v_pk_mul + 3 independent WMMAs -> fully pipelineable.
//   Phase 2: fused LN score + chunk softmax stats (m, s).
//   Phase 3: pooled partial p = w^T X via WMMA (broadcast-weight A),
//            re-reading the same 32 KB/wave tile from WGP$/L2.
// ---------------------------------------------------------------------------
__global__ __launch_bounds__(256) void ap_main(const float* __restrict__ x,
                                               float* __restrict__ ws) {
  __shared__ __align__(16) float lds_u[kD];
  __shared__ float lds_dot[kChunk], lds_sumx[kChunk], lds_ssq[kChunk];
  __shared__ float lds_wgt[kChunk], lds_red[kChunk];
  __shared__ float lds_pp[8 * kD];  // per-wave pooled partials (16 KB)

  const int t    = threadIdx.x;
  const int lane = t & 31;
  const int wave = t >> 5;      // 0..7
  const int half = lane >> 4;   // 0/1  (K half for A/B operands)
  const int lr   = lane & 15;   // row / col within 16

  const int chunk = blockIdx.x;              // 0..63
  const int batch = blockIdx.y;              // 0..15
  const int ci    = batch * kNChunk + chunk;
  const int r0    = chunk * kChunk;

  for (int j = t; j < kD; j += 256) lds_u[j] = ws[j];
  __syncthreads();

  const float sumU = ws[WS_SCAL];
  const float qb   = ws[WS_SCAL + 1];

  // ---- Phase 1: per-row reductions for this wave's 16 rows -------------
  const int rowBase = r0 + wave * 16;
  const float* xw   = x + ((size_t)batch * kN + rowBase) * kD;
  const float* xrow = xw + (size_t)lr * kD;  // B layout: lane%16 = x row

  v8f cd = {}, cs = {}, cq = {};
  v2f a_ones;
  a_ones.x = 1.0f;
  a_ones.y = 1.0f;

  for (int k0 = 0; k0 < kD; k0 += 32) {
    __builtin_prefetch(xrow + k0 + 2 * half + 64, 0, 0);
#pragma unroll
    for (int ks = 0; ks < 32; ks += 4) {
      const int kk = k0 + ks + 2 * half;     // lane>>4 picks the K half
      v2f b  = *(const v2f*)(xrow + kk);     // global_load_b64 (streams row)
      v2f au = *(const v2f*)(lds_u + kk);    // ds_load_b64, broadcast addr
      cd = __builtin_amdgcn_wmma_f32_16x16x4_f32(false, au, false, b,
                                                 (short)0, cd, false, false);
      cs = __builtin_amdgcn_wmma_f32_16x16x4_f32(false, a_ones, false, b,
                                                 (short)0, cs, false, false);
      v2f b2;
      b2.x = b.x * b.x;
      b2.y = b.y * b.y;
      cq = __builtin_amdgcn_wmma_f32_16x16x4_f32(false, a_ones, false, b2,
                                                 (short)0, cq, false, false);
    }
  }

  // Every C row identical; row 0 = VGPR0, lanes 0-15 (col = x row).
  if (half == 0) {
    lds_dot[wave * 16 + lr]  = cd[0];
    lds_sumx[wave * 16 + lr] = cs[0];
    lds_ssq[wave * 16 + lr]  = cq[0];
  }
  __syncthreads();

  // ---- Phase 2: fused LN score + chunk softmax stats -------------------
  if (t < kChunk) {
    const float dot = lds_dot[t], sx = lds_sumx[t], s2 = lds_ssq[t];
    const float mu  = sx * (1.0f / kD);
    const float var = s2 * (1.0f / kD) - mu * mu;
    const float rs  = rsqrtf(var + kEps);
    const float score = kScale * ((dot - mu * sumU) * rs + qb);
    lds_dot[t] = score;           // keep
    lds_red[t] = score;           // reduce copy
  }
  __syncthreads();
  for (int s = kChunk / 2; s >= 1; s >>= 1) {
    if (t < s) lds_red[t] = fmaxf(lds_red[t], lds_red[t + s]);
    __syncthreads();
  }
  const float m = lds_red[0];
  __syncthreads();
  if (t < kChunk) {
    const float wgt = __expf(lds_dot[t] - m);
    lds_wgt[t] = wgt;
    lds_red[t] = wgt;
  }
  __syncthreads();
  for (int s = kChunk / 2; s >= 1; s >>= 1) {
    if (t < s) lds_red[t] += lds_red[t + s];
    __syncthreads();
  }
  if (t == 0) {
    ws[WS_MS + 2 * ci]     = m;
    ws[WS_MS + 2 * ci + 1] = lds_red[0];
  }

  // ---- Phase 3: pooled partial p = w^T X via WMMA ----------------------
  // A (16x4): every row = the 4 weights of this K-chunk (broadcast rows).
  // B (4x16): rows = x rows, cols = 16 consecutive features.
  // C rows are all identical = partial pooled vector for these 16 features.
#pragma unroll 1
  for (int jb = 0; jb < kD; jb += 16) {
    v8f cc = {};
#pragma unroll
    for (int kc = 0; kc < 4; ++kc) {
      const int rA = 4 * kc + 2 * half;  // B layout: lane>>4 picks K half
      v2f bm2;
      bm2.x = xw[(size_t)rA * kD + jb + lr];        // WGP$/L2-hot re-read
      bm2.y = xw[(size_t)(rA + 1) * kD + jb + lr];
      v2f am;
      am.x = lds_wgt[wave * 16 + rA];
      am.y = lds_wgt[wave * 16 + rA + 1];
      cc = __builtin_amdgcn_wmma_f32_16x16x4_f32(false, am, false, bm2,
                                                 (short)0, cc, false, false);
    }
    if (half == 0) lds_pp[wave * kD + jb + lr] = cc[0];  // row 0 of C
  }
  __syncthreads();
  for (int j = t; j < kD; j += 256) {
    float acc = 0.0f;
#pragma unroll
    for (int w8 = 0; w8 < 8; ++w8) acc += lds_pp[w8 * kD + j];
    ws[WS_P + (size_t)ci * kD + j] = acc;
  }
}

// ---------------------------------------------------------------------------
// Kernel 3: one block per batch — merge 64 chunks with LSE rescaling.
// ---------------------------------------------------------------------------
__global__ __launch_bounds__(256) void ap_combine(const float* __restrict__ ws,
                                                  float* __restrict__ out) {
  __shared__ float mArr[kNChunk], sArr[kNChunk], fArr[kNChunk], red[kNChunk];
  const int t = threadIdx.x;
  const int b = blockIdx.x;
  const int base = b * kNChunk;

  if (t < kNChunk) {
    mArr[t] = ws[WS_MS + 2 * (base + t)];
    sArr[t] = ws[WS_MS + 2 * (base + t) + 1];
    red[t]  = mArr[t];
  }
  __syncthreads();
  for (int s = kNChunk / 2; s >= 1; s >>= 1) {
    if (t < s) red[t] = fmaxf(red[t], red[t + s]);
    __syncthreads();
  }
  const float M = red[0];
  __syncthreads();
  if (t < kNChunk) {
    const float f = __expf(mArr[t] - M);
    fArr[t] = f;
    red[t]  = sArr[t] * f;
  }
  __syncthreads();
  for (int s = kNChunk / 2; s >= 1; s >>= 1) {
    if (t < s) red[t] += red[t + s];
    __syncthreads();
  }
  const float invS = 1.0f / red[0];
  for (int j = t; j < kD; j += 256) {
    float acc = 0.0f;
    for (int ch = 0; ch < kNChunk; ++ch)
      acc += ws[WS_P + (size_t)(base + ch) * kD + j] * fArr[ch];
    out[b * kD + j] = acc * invS;
  }
}

// ---------------------------------------------------------------------------
extern "C" void kernel_launch(void* const* d_in, const int* in_sizes, int n_in,
                              void* d_out, int out_size, void* d_ws,
                              size_t ws_size, hipStream_t stream) {
  (void)in_sizes; (void)n_in; (void)out_size; (void)ws_size;
  const float* x     = (const float*)d_in[0];
  const float* query = (const float*)d_in[1];
  const float* lw    = (const float*)d_in[2];
  const float* lb    = (const float*)d_in[3];
  float* out = (float*)d_out;
  float* ws  = (float*)d_ws;

  ap_prep<<<dim3(1), dim3(kD), 0, stream>>>(query, lw, lb, ws);
  ap_main<<<dim3(kNChunk, kB), dim3(256), 0, stream>>>(x, ws);
  ap_combine<<<dim3(kB), dim3(256), 0, stream>>>(ws, out);
}